// LearnedDMemBP_69561290326258
// MI455X (gfx1250) — compile-verified
//
#include <hip/hip_runtime.h>
#include <cstdint>
#include <cstddef>

#define M_CHK   4096
#define N_VAR   8192
#define DCHK    8
#define BATCH   128
#define NITERS  10

typedef unsigned int u32x4 __attribute__((ext_vector_type(4)));
typedef int          i32x8 __attribute__((ext_vector_type(8)));
typedef int          i32x4 __attribute__((ext_vector_type(4)));

// tanh(100*v), overflow-safe: tanh(x) = sign(x) * (1 - 2/(e^{2|x|}+1))
__device__ __forceinline__ float smooth_sign(float v) {
    float a  = 100.0f * v;
    float ea = __expf(2.0f * fabsf(a));      // inf for large |a| -> t = 1
    float t  = 1.0f - 2.0f / (ea + 1.0f);
    return copysignf(t, a);
}

// llr_acc[n*B+b] = prior[n]
__global__ __launch_bounds__(256) void bp_init0(float* __restrict__ tgt,
                                                const float* __restrict__ prior) {
    int i = blockIdx.x * 256 + threadIdx.x;  // i = n*BATCH + b
    tgt[i] = prior[i >> 7];
}

// llr_acc[n*B+b] = (1-gamma[n])*prior[n] + gamma[n]*llr_old[n*B+b]
__global__ __launch_bounds__(256) void bp_initg(float* __restrict__ tgt,
                                                const float* __restrict__ src,
                                                const float* __restrict__ prior,
                                                const float* __restrict__ gamma) {
    int i = blockIdx.x * 256 + threadIdx.x;
    int n = i >> 7;
    float g = gamma[n];
    tgt[i] = fmaf(g, src[i], (1.0f - g) * prior[n]);
}

// Generic 32x32 LDS-tiled transpose: src[R,C] -> dst[C,R]
// grid = (C/32, R/32), block = (32, 8)
__global__ __launch_bounds__(256) void bp_xpose(const float* __restrict__ src,
                                                float* __restrict__ dst,
                                                int R, int C) {
    __shared__ float sm[32][33];
    int c0 = blockIdx.x * 32;
    int r0 = blockIdx.y * 32;
#pragma unroll
    for (int i = 0; i < 32; i += 8)
        sm[threadIdx.y + i][threadIdx.x] =
            src[(size_t)(r0 + threadIdx.y + i) * C + c0 + threadIdx.x];
    __syncthreads();
#pragma unroll
    for (int i = 0; i < 32; i += 8)
        dst[(size_t)(c0 + threadIdx.y + i) * R + r0 + threadIdx.x] =
            sm[threadIdx.x][threadIdx.y + i];
}

// One lane per (check m, batch b). Block = 2 checks x 128 batches.
// c2v layout:   [(m*DC+j)*BATCH + b]  (batch fastest -> coalesced)
// llr layout:   [n*BATCH + b]         (batch fastest -> coalesced gather/scatter)
// syn_t layout: [m*BATCH + b]         (coalesced)
__global__ __launch_bounds__(256) void bp_check(
    const float* __restrict__ llr_src,   // [N*B] previous llr (unused if first)
    float*       __restrict__ llr_tgt,   // [N*B] pre-initialized accumulator
    float*       __restrict__ c2v,       // [M*DC*B] read old (if !first) / write new
    const float* __restrict__ prior,     // [N]
    const int*   __restrict__ vidx,      // [M*DC]
    const float* __restrict__ syn_t,     // [M*B] transposed syndromes
    int first)
{
    __shared__ float tile[2 * DCHK * BATCH];   // 8 KB: this block's old c2v tile
    const int m  = (blockIdx.x * 256 + threadIdx.x) >> 7;
    const int b  = threadIdx.x & (BATCH - 1);
    const int m0 = blockIdx.x * 2;

    // (1) kick off the TDM DMA of the contiguous old-c2v tile first; do NOT wait yet.
    if (!first && threadIdx.x == 0) {
        unsigned long long ga =
            (unsigned long long)(uintptr_t)(c2v + (size_t)m0 * DCHK * BATCH);
        unsigned int ldsOff = (unsigned int)(uintptr_t)(&tile[0]);
        u32x4 g0;
        g0.x = 1u;                                   // count = 1 valid descriptor
        g0.y = ldsOff;                               // lds_addr
        g0.z = (unsigned int)(ga & 0xFFFFFFFFull);   // global_addr[31:0]
        g0.w = (unsigned int)((ga >> 32) & 0x01FFFFFFull) | (2u << 30); // addr[56:32] | type=2
        i32x8 g1;
        g1[0] = (int)(2u << 16);        // data_size = 2 (4 bytes/elem)
        g1[1] = (int)(2048u << 16);     // tensor_dim0[15:0] = 2048 (bits 63:48)
        g1[2] = (int)(1u << 16);        // tensor_dim0 hi = 0, tensor_dim1 = 1
        g1[3] = (int)(2048u << 16);     // tile_dim0 = 2048 (bits 127:112)
        g1[4] = 1;                      // tile_dim1 = 1
        g1[5] = 2048;                   // tensor_dim0_stride = 2048
        g1[6] = 0;
        g1[7] = 0;
        i32x4 g2 = {1, 1, 0, 0};        // tensor_dim2 = 1, tensor_dim3 = 1
        i32x4 g3 = {0, 0, 0, 0};
        i32x8 g4 = {0, 0, 0, 0, 0, 0, 0, 0};  // extra group (clang-23 6-arg form)
        __builtin_amdgcn_tensor_load_to_lds(g0, g1, g2, g3, g4, 0);
    }

    // (2) overlap with the DMA: wave-uniform vidx row (1 cacheline/wave), syndrome,
    //     and all 8 independent random llr gathers in flight per lane.
    const int ebase = m * DCHK;
    int idxs[DCHK];
#pragma unroll
    for (int j = 0; j < DCHK; ++j) idxs[j] = vidx[ebase + j];

    const float ss = 1.0f - 2.0f * syn_t[m * BATCH + b];

    float v[DCHK];
    if (first) {
#pragma unroll
        for (int j = 0; j < DCHK; ++j) v[j] = prior[idxs[j]];
    } else {
        float g[DCHK];
#pragma unroll
        for (int j = 0; j < DCHK; ++j) g[j] = llr_src[idxs[j] * BATCH + b];
        // (3) now resolve the DMA and combine.
        if (threadIdx.x == 0) __builtin_amdgcn_s_wait_tensorcnt(0);
        __syncthreads();
        const int lbase = (m & 1) * DCHK * BATCH + b;
#pragma unroll
        for (int j = 0; j < DCHK; ++j) v[j] = g[j] - tile[lbase + j * BATCH];
    }

    float s[DCHK], ab[DCHK];
#pragma unroll
    for (int j = 0; j < DCHK; ++j) { s[j] = smooth_sign(v[j]); ab[j] = fabsf(v[j]); }

    // exclusive sign product via prefix/suffix (no divide-by-~0 hazard)
    float excl[DCHK];
    {
        float pre = 1.0f;
#pragma unroll
        for (int j = 0; j < DCHK; ++j) { excl[j] = pre; pre *= s[j]; }
        float suf = 1.0f;
#pragma unroll
        for (int j = DCHK - 1; j >= 0; --j) { excl[j] *= suf; suf *= s[j]; }
    }

    // row min / second-min / argmin of |v|
    float m1 = 1e30f, m2v = 1e30f; int jmin = 0;
#pragma unroll
    for (int j = 0; j < DCHK; ++j) {
        if (ab[j] < m1)       { m2v = m1; m1 = ab[j]; jmin = j; }
        else if (ab[j] < m2v) { m2v = ab[j]; }
    }

    // softmin weights, shifted by global row min (ratio is shift-invariant)
    float w[DCHK], S1 = 0.0f, S2 = 0.0f;
#pragma unroll
    for (int j = 0; j < DCHK; ++j) {
        w[j] = __expf((m1 - ab[j]) * 100.0f);   // arg <= 0, underflows safely
        S1  += w[j];
        S2  += ab[j] * w[j];
    }

#pragma unroll
    for (int j = 0; j < DCHK; ++j) {
        float me;
        if (j == jmin) {
            // argmin row: re-shift by second-min so denominator contains weight 1.0
            float T1 = 0.0f, T2 = 0.0f;
#pragma unroll
            for (int k = 0; k < DCHK; ++k) {
                if (k == j) continue;
                float ww = __expf((m2v - ab[k]) * 100.0f);
                T1 += ww;
                T2 += ab[k] * ww;
            }
            me = T2 / T1;
        } else {
            me = (S2 - ab[j] * w[j]) / (S1 - w[j]);   // denominator >= w[jmin] = 1
        }
        float c = ss * excl[j] * me;
        c2v[(ebase + j) * BATCH + b] = c;
        atomicAdd(&llr_tgt[idxs[j] * BATCH + b], c);  // global_atomic_add_f32
    }
}

extern "C" void kernel_launch(void* const* d_in, const int* in_sizes, int n_in,
                              void* d_out, int out_size, void* d_ws, size_t ws_size,
                              hipStream_t stream) {
    const float* syn   = (const float*)d_in[0];   // [B*M]
    const float* prior = (const float*)d_in[1];   // [N]
    const float* gamma = (const float*)d_in[2];   // [N]
    const int*   vidx  = (const int*)d_in[3];     // [M*DC]
    float* out = (float*)d_out;                   // [B*N]

    char* ws = (char*)d_ws;
    const size_t llrBytes = (size_t)N_VAR * BATCH * sizeof(float);        // 4 MB
    const size_t c2vBytes = (size_t)M_CHK * DCHK * BATCH * sizeof(float); // 16 MB
    float* llrA  = (float*)(ws);
    float* llrB  = (float*)(ws + llrBytes);
    float* c2v   = (float*)(ws + 2 * llrBytes);
    float* syn_t = (float*)(ws + 2 * llrBytes + c2vBytes);                // 2 MB

    const int initBlocks  = (N_VAR * BATCH) / 256;   // 4096
    const int checkBlocks = (M_CHK * BATCH) / 256;   // 2048

    // one-time (per launch): syndromes [B,M] -> [M,B] so bp_check reads coalesce
    bp_xpose<<<dim3(M_CHK / 32, BATCH / 32), dim3(32, 8), 0, stream>>>(
        syn, syn_t, BATCH, M_CHK);

    // iteration 0: v2c = prior on every edge; llr = segsum + prior
    bp_init0<<<initBlocks, 256, 0, stream>>>(llrA, prior);
    bp_check<<<checkBlocks, 256, 0, stream>>>(llrA, llrA, c2v, prior, vidx, syn_t, 1);

    float* src = llrA;
    float* tgt = llrB;
    for (int it = 1; it < NITERS; ++it) {
        bp_initg<<<initBlocks, 256, 0, stream>>>(tgt, src, prior, gamma);
        bp_check<<<checkBlocks, 256, 0, stream>>>(src, tgt, c2v, prior, vidx, syn_t, 0);
        float* t = src; src = tgt; tgt = t;
    }
    // final llr (internal [N,B]) -> reference output [B,N]
    bp_xpose<<<dim3(BATCH / 32, N_VAR / 32), dim3(32, 8), 0, stream>>>(
        src, out, N_VAR, BATCH);
}